// GNNLayer_16561393893518
// MI455X (gfx1250) — compile-verified
//
#include <hip/hip_runtime.h>
#include <hip/hip_bf16.h>

typedef __attribute__((ext_vector_type(2))) float v2f;
typedef __attribute__((ext_vector_type(8))) float v8f;

#define D_FEAT 64   // D_IN == D_OUT == 64

// Guaranteed-hardware f32 atomic add (GLOBAL_ATOMIC_ADD_F32, no return -> STOREcnt).
__device__ __forceinline__ void atomic_add_f32(float* addr, float val) {
    asm volatile("global_atomic_add_f32 %0, %1, off" :: "v"(addr), "v"(val) : "memory");
}

// K1: zero the aggregation buffer, set deg = 1 (self-loop included).
__global__ void gcn_init(float* __restrict__ agg, float* __restrict__ deg, int N) {
    int t = blockIdx.x * blockDim.x + threadIdx.x;
    if (t < N * D_FEAT) agg[t] = 0.0f;
    if (t < N)          deg[t] = 1.0f;
}

// K2: degree over destination nodes (one thread per edge).
__global__ void gcn_degree(const int* __restrict__ ei, float* __restrict__ deg, int E) {
    int t = blockIdx.x * blockDim.x + threadIdx.x;
    if (t < E) {
        int d = ei[E + t];          // edge_index[1][t]
        atomic_add_f32(&deg[d], 1.0f);
    }
}

// K3: dinv = rsqrt(deg)   (deg >= 1 always).
__global__ void gcn_rsqrt(const float* __restrict__ deg, float* __restrict__ dinv, int N) {
    int t = blockIdx.x * blockDim.x + threadIdx.x;
    if (t < N) dinv[t] = rsqrtf(deg[t]);
}

// K4: H = X @ W with V_WMMA_F32_16X16X4_F32. One wave per 16x16 output tile.
// K = 64 -> 16 chained WMMA steps. All branching is wave-uniform (EXEC all-ones).
//
// f32 WMMA VGPR layouts (ISA 7.12.2):
//   A 16x4 : lane L holds row m = L%16; vgpr j holds K = (L/16)*2 + j
//   B 4x16 : lane L holds col n = L%16; vgpr j holds K = (L/16)*2 + j
//   C 16x16: lane L holds col n = L%16; vgpr j holds row m = j + (L/16)*8
__global__ void gcn_gemm_wmma(const float* __restrict__ X, const float* __restrict__ W,
                              float* __restrict__ H, int nTiles) {
    const int lane = threadIdx.x & 31;
    const int wave = threadIdx.x >> 5;
    const int tile = blockIdx.x * 8 + wave;          // 8 waves / block
    if (tile >= nTiles) return;                      // wave-uniform guard

    const int rowTile = tile >> 2;                   // 4 col-tiles across D_OUT=64
    const int colTile = tile & 3;

    const int m     = lane & 15;
    const int khalf = (lane >> 4) << 1;              // 0 or 2
    const int col   = colTile * 16 + (lane & 15);
    const size_t rowBase = (size_t)(rowTile * 16 + m) * D_FEAT;

    v8f c = {};
#pragma unroll
    for (int kk = 0; kk < 16; ++kk) {
        const int k0 = kk * 4 + khalf;
        v2f a, b;
        a.x = X[rowBase + k0];                       // contiguous pair -> b64 load
        a.y = X[rowBase + k0 + 1];
        b.x = W[(size_t)k0 * D_FEAT + col];          // W is [64(in), 64(out)] row-major
        b.y = W[(size_t)(k0 + 1) * D_FEAT + col];
        c = __builtin_amdgcn_wmma_f32_16x16x4_f32(
                /*neg_a=*/false, a, /*neg_b=*/false, b,
                /*c_mod=*/(short)0, c, /*reuse_a=*/false, /*reuse_b=*/false);
    }

    const int mhi = (lane >> 4) << 3;                // 0 or 8
#pragma unroll
    for (int j = 0; j < 8; ++j) {
        const int mm = mhi + j;
        H[(size_t)(rowTile * 16 + mm) * D_FEAT + col] = c[j];
    }
}

// K5: edge scatter. One wave per edge; lane l owns features {2l, 2l+1}.
// Gather of h[src] is one coalesced b64 per lane; 2 hardware f32 atomics per lane.
__global__ void gcn_edge_scatter(const int* __restrict__ ei, const float* __restrict__ H,
                                 const float* __restrict__ dinv, float* __restrict__ agg,
                                 int E) {
    const int gw   = (int)((blockIdx.x * (size_t)blockDim.x + threadIdx.x) >> 5);
    const int lane = threadIdx.x & 31;
    if (gw >= E) return;                             // wave-uniform

    const int s = ei[gw];                            // edge_index[0][e]
    const int d = ei[E + gw];                        // edge_index[1][e]
    const float nrm = dinv[s] * dinv[d];

    const v2f v = ((const v2f*)(H + (size_t)s * D_FEAT))[lane];
    float* ap = agg + (size_t)d * D_FEAT + (lane << 1);
    atomic_add_f32(ap,     v.x * nrm);
    atomic_add_f32(ap + 1, v.y * nrm);
}

// K6: out = relu(agg + h*dinv^2 (self-loop msg) + b)
__global__ void gcn_finalize(const float* __restrict__ agg, const float* __restrict__ H,
                             const float* __restrict__ dinv, const float* __restrict__ bias,
                             float* __restrict__ out, int N) {
    int t = blockIdx.x * blockDim.x + threadIdx.x;
    if (t >= N * D_FEAT) return;
    const int node = t >> 6;
    const int f    = t & 63;
    const float di = dinv[node];
    const float v  = agg[t] + H[t] * (di * di) + bias[f];
    out[t] = fmaxf(v, 0.0f);
}

extern "C" void kernel_launch(void* const* d_in, const int* in_sizes, int n_in,
                              void* d_out, int out_size, void* d_ws, size_t ws_size,
                              hipStream_t stream) {
    const float* x    = (const float*)d_in[0];   // [N, 64]
    const int*   ei   = (const int*)d_in[1];     // [2, E]
    const float* W    = (const float*)d_in[2];   // [64, 64]
    const float* bias = (const float*)d_in[3];   // [64]
    float* out = (float*)d_out;

    const int N = in_sizes[0] / D_FEAT;          // 80000
    const int E = in_sizes[1] / 2;               // 1280000

    // Workspace layout: h [N*64] | agg [N*64] | deg [N] | dinv [N]
    float* h    = (float*)d_ws;
    float* agg  = h   + (size_t)N * D_FEAT;
    float* deg  = agg + (size_t)N * D_FEAT;
    float* dinv = deg + N;

    const int nd = N * D_FEAT;

    gcn_init<<<(nd + 255) / 256, 256, 0, stream>>>(agg, deg, N);
    gcn_degree<<<(E + 255) / 256, 256, 0, stream>>>(ei, deg, E);
    gcn_rsqrt<<<(N + 255) / 256, 256, 0, stream>>>(deg, dinv, N);

    const int nTiles = (N / 16) * 4;             // 20000 16x16 tiles
    gcn_gemm_wmma<<<(nTiles + 7) / 8, 256, 0, stream>>>(x, W, h, nTiles);

    // one wave (32 lanes) per edge -> 8 edges per 256-thread block
    gcn_edge_scatter<<<(E + 7) / 8, 256, 0, stream>>>(ei, h, dinv, agg, E);

    gcn_finalize<<<(nd + 255) / 256, 256, 0, stream>>>(agg, h, dinv, bias, out, N);
}